// SqueezeAndExciteFusionAdd_10299331576156
// MI455X (gfx1250) — compile-verified
//
#include <hip/hip_runtime.h>
#include <math.h>

// Shapes from the reference
#define BB   16          // batch
#define CC   512         // channels
#define HHID 32          // hidden (C/16)
#define HW   4096        // 64*64 spatial
#define PER_BRANCH (BB * CC)   // 8192 pooled/gate values per branch

typedef __attribute__((ext_vector_type(2))) float v2f;
typedef __attribute__((ext_vector_type(8))) float v8f;

// ---------------------------------------------------------------------------
// Kernel 1: spatial mean pooling. One 256-thread block per (branch, b, c);
// each thread streams float4s (global_load_b128) and we LDS-tree-reduce.
// Pure bandwidth: reads 2 x 128 MiB.
// ---------------------------------------------------------------------------
__global__ void pool_kernel(const float* __restrict__ rgb,
                            const float* __restrict__ depth,
                            float* __restrict__ pooled /* [2][16][512] */) {
    const int id = blockIdx.x;                       // [0, 2*8192)
    const int bc = id & (PER_BRANCH - 1);
    const float* src = (id >= PER_BRANCH) ? depth : rgb;
    const float4* p = (const float4*)(src + (size_t)bc * HW);

    float s = 0.f;
    for (int i = threadIdx.x; i < HW / 4; i += 256) {
        float4 v = p[i];
        s += (v.x + v.y) + (v.z + v.w);
    }

    __shared__ float red[256];
    red[threadIdx.x] = s;
    __syncthreads();
    for (int off = 128; off > 0; off >>= 1) {
        if (threadIdx.x < off) red[threadIdx.x] += red[threadIdx.x + off];
        __syncthreads();
    }
    if (threadIdx.x == 0) pooled[id] = red[0] * (1.0f / (float)HW);
}

// ---------------------------------------------------------------------------
// Kernel 2: SE MLP for one branch, in fp32 WMMA (V_WMMA_F32_16X16X4_F32).
// GEMM1: h1[16x32] = relu(pooled[16x512] @ w1^T[512x32] + b1)   (128 k-steps)
// GEMM2: gate[16x512] = sigmoid(h1[16x32] @ w2^T[32x512] + b2)  (8 k-steps)
// One block, 8 waves. Waves 0-1 do the two GEMM1 N-tiles; all 8 waves split
// the 32 GEMM2 N-tiles. A/B layouts per CDNA5 ISA 7.12.2 (32-bit 16x4).
// ---------------------------------------------------------------------------
__global__ void gate_kernel(const float* __restrict__ pooled, // [16][512] this branch
                            const float* __restrict__ w1,     // [32][512]
                            const float* __restrict__ b1,     // [32]
                            const float* __restrict__ w2,     // [512][32]
                            const float* __restrict__ b2,     // [512]
                            float* __restrict__ gates) {      // [16][512] this branch
    __shared__ float h1[BB][HHID];

    const int lane  = threadIdx.x & 31;
    const int wave  = threadIdx.x >> 5;   // 0..7
    const int row   = lane & 15;          // M (A) or N (B) index within tile
    const int khalf = lane >> 4;          // 0: K={0,1}, 1: K={2,3}

    // ---- GEMM1 ----
    if (wave < 2) {
        const int nbase = wave * 16;
        const float* Arow = pooled + row * CC;            // pooled[b=row][:]
        const float* Brow = w1 + (nbase + row) * CC;      // w1[h=N][:] (B = w1^T)
        v8f acc = {0.f, 0.f, 0.f, 0.f, 0.f, 0.f, 0.f, 0.f};
        for (int k = 0; k < CC; k += 4) {
            v2f av = *(const v2f*)(Arow + k + khalf * 2);
            v2f bv = *(const v2f*)(Brow + k + khalf * 2);
            acc = __builtin_amdgcn_wmma_f32_16x16x4_f32(
                false, av, false, bv, (short)0, acc, false, false);
        }
        const float bias = b1[nbase + row];
        #pragma unroll
        for (int r = 0; r < 8; ++r) {
            const int m = r + khalf * 8;                  // batch index
            float v = acc[r] + bias;
            h1[m][nbase + row] = v > 0.f ? v : 0.f;       // ReLU
        }
    }
    __syncthreads();

    // ---- GEMM2 ----
    for (int t = wave; t < CC / 16; t += 8) {
        const int n = t * 16 + row;                       // output channel
        const float* Brow = w2 + n * HHID;                // w2[c=N][:] (B = w2^T)
        v8f acc = {0.f, 0.f, 0.f, 0.f, 0.f, 0.f, 0.f, 0.f};
        #pragma unroll
        for (int k = 0; k < HHID; k += 4) {
            v2f av;
            av.x = h1[row][k + khalf * 2];
            av.y = h1[row][k + khalf * 2 + 1];
            v2f bv = *(const v2f*)(Brow + k + khalf * 2);
            acc = __builtin_amdgcn_wmma_f32_16x16x4_f32(
                false, av, false, bv, (short)0, acc, false, false);
        }
        const float bias = b2[n];
        #pragma unroll
        for (int r = 0; r < 8; ++r) {
            const int m = r + khalf * 8;                  // batch index
            const float pre = acc[r] + bias;
            gates[m * CC + n] = 1.0f / (1.0f + __expf(-pre)); // sigmoid
        }
    }
}

// ---------------------------------------------------------------------------
// Kernel 3: out = rgb*gate_r + depth*gate_d, float4 vectorized streaming.
// Reads 2 x 128 MiB, writes 128 MiB. 1024 consecutive threads share one gate
// value -> gate loads are broadcast hits in cache.
// ---------------------------------------------------------------------------
__global__ void fuse_kernel(const float* __restrict__ rgb,
                            const float* __restrict__ depth,
                            const float* __restrict__ gates,  // [2][16][512]
                            float* __restrict__ out) {
    const size_t i  = (size_t)blockIdx.x * blockDim.x + threadIdx.x; // float4 idx
    const size_t bc = i >> 10;                                       // /(4096/4)
    const float gr = gates[bc];
    const float gd = gates[PER_BRANCH + bc];
    const float4 r = ((const float4*)rgb)[i];
    const float4 d = ((const float4*)depth)[i];
    float4 o;
    o.x = fmaf(r.x, gr, d.x * gd);
    o.y = fmaf(r.y, gr, d.y * gd);
    o.z = fmaf(r.z, gr, d.z * gd);
    o.w = fmaf(r.w, gr, d.w * gd);
    ((float4*)out)[i] = o;
}

extern "C" void kernel_launch(void* const* d_in, const int* in_sizes, int n_in,
                              void* d_out, int out_size, void* d_ws, size_t ws_size,
                              hipStream_t stream) {
    const float* rgb   = (const float*)d_in[0];
    const float* depth = (const float*)d_in[1];
    const float* w1r   = (const float*)d_in[2];
    const float* b1r   = (const float*)d_in[3];
    const float* w2r   = (const float*)d_in[4];
    const float* b2r   = (const float*)d_in[5];
    const float* w1d   = (const float*)d_in[6];
    const float* b1d   = (const float*)d_in[7];
    const float* w2d   = (const float*)d_in[8];
    const float* b2d   = (const float*)d_in[9];

    float* pooled = (float*)d_ws;                 // [2][16][512] = 64 KiB
    float* gates  = pooled + 2 * PER_BRANCH;      // [2][16][512] = 64 KiB
    float* out    = (float*)d_out;

    // 1) mean pool both tensors
    pool_kernel<<<2 * PER_BRANCH, 256, 0, stream>>>(rgb, depth, pooled);

    // 2) SE MLPs (fp32 WMMA), one tiny launch per branch
    gate_kernel<<<1, 256, 0, stream>>>(pooled, w1r, b1r, w2r, b2r, gates);
    gate_kernel<<<1, 256, 0, stream>>>(pooled + PER_BRANCH, w1d, b1d, w2d, b2d,
                                       gates + PER_BRANCH);

    // 3) gated fuse-add, float4 streaming
    const size_t n4 = (size_t)BB * CC * HW / 4;   // 8,388,608 float4s
    fuse_kernel<<<(unsigned)(n4 / 256), 256, 0, stream>>>(rgb, depth, gates, out);
}